// ChamferDistance_63127429316928
// MI455X (gfx1250) — compile-verified
//
#include <hip/hip_runtime.h>

typedef __attribute__((ext_vector_type(2))) float v2f;
typedef __attribute__((ext_vector_type(8))) float v8f;

// One wave owns 64 x-points (4 B-tiles of 16), streams all y-tiles.
#define XS_PER_WAVE   64
#define WAVES_PER_BLK 8
#define XS_PER_BLK    (XS_PER_WAVE * WAVES_PER_BLK)   // 512

__global__ void chamfer_zero_kernel(float* out) { out[0] = 0.0f; }

// Unsigned min: for non-negative IEEE floats, u32 ordering == float ordering.
// Lowers to bare v_min_u32 (no NaN canonicalization, freely schedulable).
__device__ __forceinline__ unsigned umin32(unsigned a, unsigned b) {
    return a < b ? a : b;
}

__global__ __launch_bounds__(256)
void chamfer_dir_kernel(const float4* __restrict__ X,   // points we find mins FOR
                        const float4* __restrict__ Y,   // points we search OVER
                        float* __restrict__ out,
                        int y_tiles,                    // npts / 16
                        float inv_n)                    // 1 / npts
{
    const int wave  = threadIdx.x >> 5;
    const int lane  = threadIdx.x & 31;
    const int n     = lane & 15;          // point-in-tile index
    const int hi    = lane >> 4;          // 0: K={0,1}, 1: K={2,3}
    const int xbase = blockIdx.x * XS_PER_BLK + wave * XS_PER_WAVE;

    // ---- Load 4 resident B-tiles (x side): B[k][n] = (x, y, z, 1) ----
    // C operand carries the per-column |x|^2 so the WMMA emits the full
    // squared distance: D[m][n] = |x_n|^2 + |y_m|^2 - 2*(y_m . x_n)  >= 0.
    v2f b[4];
    v8f c[4];
#pragma unroll
    for (int s = 0; s < 4; ++s) {
        float4 xp = X[xbase + s * 16 + n];
        float x2  = xp.x * xp.x + xp.y * xp.y + xp.z * xp.z;
        b[s].x = hi ? xp.z : xp.x;        // K2 : K0
        b[s].y = hi ? 1.0f : xp.y;        // K3 : K1
#pragma unroll
        for (int r = 0; r < 8; ++r) c[s][r] = x2;   // per-column constant
    }

    // ---- Running mins as uint bit-patterns of non-negative floats ----
    unsigned m[4][8];
#pragma unroll
    for (int s = 0; s < 4; ++s)
#pragma unroll
        for (int r = 0; r < 8; ++r)
            m[s][r] = 0x7F7FFFFFu;        // bits of FLT_MAX

    // ---- Stream all y-tiles ----
#pragma unroll 2
    for (int t = 0; t < y_tiles; ++t) {
        float4 yp = Y[t * 16 + n];
        float  r2 = yp.x * yp.x + yp.y * yp.y + yp.z * yp.z;
        v2f a;
        a.x = hi ? (-2.0f * yp.z) : (-2.0f * yp.x);   // K2 : K0
        a.y = hi ? r2             : (-2.0f * yp.y);   // K3 : K1

#pragma unroll
        for (int s = 0; s < 4; ++s) {
            v8f d = __builtin_amdgcn_wmma_f32_16x16x4_f32(
                false, a, false, b[s], (short)0, c[s], false, false);
#pragma unroll
            for (int r = 0; r < 8; ++r)
                m[s][r] = umin32(m[s][r], __float_as_uint(d[r]));
        }
    }

    // ---- Reduce: 8 rows in-lane, merge lane halves (|x|^2 already inside) ----
    float partial = 0.0f;
#pragma unroll
    for (int s = 0; s < 4; ++s) {
        unsigned v = m[s][0];
#pragma unroll
        for (int r = 1; r < 8; ++r) v = umin32(v, m[s][r]);
        v = umin32(v, (unsigned)__shfl_xor((int)v, 16, 32)); // merge M=0..7 / M=8..15
        partial += __uint_as_float(v);                       // duplicated across halves
    }
    // Sum the 16 distinct values (both halves hold identical copies).
    partial += __shfl_xor(partial, 1, 32);
    partial += __shfl_xor(partial, 2, 32);
    partial += __shfl_xor(partial, 4, 32);
    partial += __shfl_xor(partial, 8, 32);

    if (lane == 0)
        atomicAdd(out, partial * inv_n);
}

extern "C" void kernel_launch(void* const* d_in, const int* in_sizes, int n_in,
                              void* d_out, int out_size, void* d_ws, size_t ws_size,
                              hipStream_t stream) {
    (void)n_in; (void)out_size; (void)d_ws; (void)ws_size;

    const float4* pred = (const float4*)d_in[0];
    const float4* targ = (const float4*)d_in[1];
    float* out = (float*)d_out;

    const int npts    = in_sizes[0] / 4;     // 131072
    const int y_tiles = npts / 16;           // 8192
    const int blocks  = npts / XS_PER_BLK;   // 256 per direction
    const float inv_n = 1.0f / (float)npts;

    hipLaunchKernelGGL(chamfer_zero_kernel, dim3(1), dim3(1), 0, stream, out);
    hipLaunchKernelGGL(chamfer_dir_kernel, dim3(blocks), dim3(256), 0, stream,
                       pred, targ, out, y_tiles, inv_n);   // p -> t
    hipLaunchKernelGGL(chamfer_dir_kernel, dim3(blocks), dim3(256), 0, stream,
                       targ, pred, out, y_tiles, inv_n);   // t -> p
}